// GraphConvDAGPoolNN_83854941487717
// MI455X (gfx1250) — compile-verified
//
#include <hip/hip_runtime.h>
#include <math.h>

// ---------------------------------------------------------------------------
// GraphConvDAGPoolNN for MI455X (gfx1250), wave32 + V_WMMA_F32_16X16X4_F32.
// Only the live subgraph is computed (pool branch is dead in the reference).
// ---------------------------------------------------------------------------

typedef __attribute__((ext_vector_type(2))) float v2f;
typedef __attribute__((ext_vector_type(8))) float v8f;

#define KC 68   // LDS K-chunk (transposed stage, 34816 B); 136 = 2*68, 8 < 68

// C[M,128] = A[M,K] @ B[K,128], A row stride == K (true at all call sites).
// Block: 256 threads = 8 waves; each wave: 16 rows x 128 cols (8 acc tiles).
// EPI: fused bias + relu epilogue.
template <int K, bool EPI>
__global__ __launch_bounds__(256)
void gemm_wmma_f32(const float* __restrict__ A,
                   const float* __restrict__ B,     // [K,128] row-major
                   const float* __restrict__ bias,  // used iff EPI
                   float* __restrict__ C, int M)
{
    constexpr int KCC    = (K < KC) ? K : KC;       // 8 or 68
    constexpr int NCHUNK = K / KCC;                 // 1 or 2 (exact)

    __shared__ float sB[128 * KC];                  // transposed: sB[col*KC + k]

    const int tid  = threadIdx.x;
    const int wave = tid >> 5;
    const int lane = tid & 31;
    const int half = lane >> 4;                     // lanes 16-31: K+2 / M+8
    const int l    = lane & 15;

    const int row  = blockIdx.x * 128 + wave * 16 + l;
    // Clamp instead of branching: OOB tile rows compute garbage but are
    // never stored (epilogue guards). Keeps EXEC all-1s for WMMA.
    const int rowc = (row < M) ? row : (M - 1);
    const float* Arow = A + (size_t)rowc * K;

    v8f acc[8];
#pragma unroll
    for (int t = 0; t < 8; ++t) acc[t] = (v8f)0.0f;

#pragma unroll
    for (int chunk = 0; chunk < NCHUNK; ++chunk) {
        const int kb = chunk * KCC;

        // Cooperative transposed stage of B chunk into LDS.
#pragma unroll
        for (int idx = tid; idx < KCC * 128; idx += 256) {
            const int k = idx >> 7, col = idx & 127;
            sB[col * KC + k] = B[(size_t)(kb + k) * 128 + col];
        }
        __syncthreads();

        // Preload this lane's A slice for the whole chunk (batched b64 loads;
        // 16x4 f32 layout: VGPR0 holds K=k0+2*half, VGPR1 holds K=k0+2*half+1).
        v2f aReg[KCC / 4];
#pragma unroll
        for (int j = 0; j < KCC / 4; ++j)
            aReg[j] = *(const v2f*)(Arow + kb + j * 4 + 2 * half);

#pragma unroll
        for (int k0 = 0; k0 < KCC; k0 += 4) {
            // Batch the 8 B fragments first so DS waits overlap WMMA issue.
            v2f b[8];
#pragma unroll
            for (int t = 0; t < 8; ++t)
                b[t] = *(const v2f*)(&sB[(t * 16 + l) * KC + k0 + 2 * half]);
#pragma unroll
            for (int t = 0; t < 8; ++t)
                acc[t] = __builtin_amdgcn_wmma_f32_16x16x4_f32(
                    false, aReg[k0 / 4], false, b[t], (short)0, acc[t],
                    false, false);
        }
        __syncthreads();
    }

    // Epilogue: VGPR i of C tile holds row (i + 8*half), col = l.
    const int rbase = blockIdx.x * 128 + wave * 16 + 8 * half;
    // Wave-uniform full-tile test: all but the last block take the unguarded
    // path (scalar branch, no per-element exec masking).
    const bool fullTile = (blockIdx.x * 128 + 128) <= M;
#pragma unroll
    for (int t = 0; t < 8; ++t) {
        const int col = t * 16 + l;
        float* Ccol = C + (size_t)rbase * 128 + col;
        const float bv = EPI ? bias[col] : 0.0f;
        if (fullTile) {
#pragma unroll
            for (int i = 0; i < 8; ++i) {
                float v = acc[t][i];
                if (EPI) v = fmaxf(v + bv, 0.0f);
                Ccol[(size_t)i * 128] = v;
            }
        } else {
#pragma unroll
            for (int i = 0; i < 8; ++i) {
                if (rbase + i < M) {
                    float v = acc[t][i];
                    if (EPI) v = fmaxf(v + bv, 0.0f);
                    Ccol[(size_t)i * 128] = v;
                }
            }
        }
    }
}

// ----------------------- degree / normalization ----------------------------
__global__ void deg_init(float* __restrict__ deg, int n)
{
    int v = blockIdx.x * blockDim.x + threadIdx.x;
    if (v < n) deg[v] = 1.0f;                     // self-loop
}

__global__ void deg_accum(const long long* __restrict__ ei, int E,
                          float* __restrict__ deg)
{
    int e = blockIdx.x * blockDim.x + threadIdx.x;
    if (e < E) atomicAdd(&deg[ei[2ll * e + 1]], 1.0f);   // dst column
}

__global__ void deg_rsqrt(float* __restrict__ deg, int n)
{
    int v = blockIdx.x * blockDim.x + threadIdx.x;
    if (v < n) deg[v] = rsqrtf(deg[v]);
}

// --------------------------- edge scatter ----------------------------------
// One wave per edge: acc[d,:] += xw[s,:] * dinv[s]*dinv[d]   (128 feats).
// Edge id is wave-uniform -> readfirstlane so endpoints/norm go scalar (SMEM).
__global__ __launch_bounds__(256)
void gcn_scatter(const long long* __restrict__ ei, int E,
                 const float* __restrict__ dinv,
                 const float* __restrict__ xw,
                 float* __restrict__ acc)
{
    const int e = __builtin_amdgcn_readfirstlane(blockIdx.x * 8 + (threadIdx.x >> 5));
    if (e >= E) return;
    const int lane = threadIdx.x & 31;
    const long long s = ei[2ll * e];
    const long long d = ei[2ll * e + 1];
    const float norm = dinv[s] * dinv[d];

    const float4 v = ((const float4*)(xw + (size_t)s * 128))[lane];
    float* dst = acc + (size_t)d * 128 + lane * 4;
    atomicAdd(dst + 0, v.x * norm);
    atomicAdd(dst + 1, v.y * norm);
    atomicAdd(dst + 2, v.z * norm);
    atomicAdd(dst + 3, v.w * norm);
}

// ---------------- finalize (self-loop + bias + relu) and concat ------------
// h = relu(acc + xw*dinv^2 + bias);  z = xFirst ? [x|h] : [h|x]   ([N,136])
__global__ __launch_bounds__(256)
void finalize_concat(const float* __restrict__ acc, const float* __restrict__ xw,
                     const float* __restrict__ dinv, const float* __restrict__ bias,
                     const float* __restrict__ x,   // [N,8]
                     float* __restrict__ z, int N, int xFirst)
{
    const int v = blockIdx.x * 2 + (threadIdx.x >> 7);
    if (v >= N) return;
    const int col = threadIdx.x & 127;
    const float di = dinv[v];
    float h = acc[(size_t)v * 128 + col]
            + xw[(size_t)v * 128 + col] * di * di + bias[col];
    h = fmaxf(h, 0.0f);
    float* zr = z + (size_t)v * 136;
    if (xFirst) {
        zr[8 + col] = h;
        if (col < 8) zr[col] = x[(size_t)v * 8 + col];
    } else {
        zr[col] = h;
        if (col < 8) zr[128 + col] = x[(size_t)v * 8 + col];
    }
}

// -------- final head: sigmoid([x|h4] @ f5W + f5b), f5W is [136,1] ----------
__global__ __launch_bounds__(256)
void final_out(const float* __restrict__ x, const float* __restrict__ h4,
               const float* __restrict__ f5W, const float* __restrict__ f5b,
               float* __restrict__ out, int N)
{
    const int v = blockIdx.x * blockDim.x + threadIdx.x;
    if (v >= N) return;
    float s = f5b[0];
    const float4* xr = (const float4*)(x + (size_t)v * 8);
    const float4* wx = (const float4*)f5W;
#pragma unroll
    for (int i = 0; i < 2; ++i) {
        float4 a = xr[i], w = wx[i];
        s += a.x * w.x + a.y * w.y + a.z * w.z + a.w * w.w;
    }
    const float4* hr = (const float4*)(h4 + (size_t)v * 128);
    const float4* wh = (const float4*)(f5W + 8);
#pragma unroll 8
    for (int i = 0; i < 32; ++i) {
        float4 a = hr[i], w = wh[i];
        s += a.x * w.x + a.y * w.y + a.z * w.z + a.w * w.w;
    }
    out[v] = 1.0f / (1.0f + __expf(-s));
}

// ---------------------------------------------------------------------------
extern "C" void kernel_launch(void* const* d_in, const int* in_sizes, int n_in,
                              void* d_out, int out_size, void* d_ws, size_t ws_size,
                              hipStream_t stream)
{
    const float*     x   = (const float*)d_in[0];
    const long long* ei  = (const long long*)d_in[1];   // int64 [E,2]
    const float*     c1W = (const float*)d_in[2];
    const float*     c1b = (const float*)d_in[3];
    const float*     c2W = (const float*)d_in[4];
    const float*     c2b = (const float*)d_in[5];
    const float*     f4W = (const float*)d_in[18];
    const float*     f4b = (const float*)d_in[19];
    const float*     f5W = (const float*)d_in[20];
    const float*     f5b = (const float*)d_in[21];
    float* out = (float*)d_out;

    const int N = in_sizes[0] / 8;
    const int E = in_sizes[1] / 2;

    // Workspace layout (floats): [deg | bufA(N*128) | bufB(N*128) | bufC(N*136)]
    float* ws = (float*)d_ws;
    const size_t nAlign = (((size_t)N) + 1023) & ~(size_t)1023;
    float* deg  = ws;
    float* bufA = ws + nAlign;                       // xw1 / xw2 / h4
    float* bufB = bufA + (size_t)N * 128;            // scatter accumulators
    float* bufC = bufB + (size_t)N * 128;            // z / xr  [N,136]

    const int gN  = (N + 255) / 256;
    const int gE  = (E + 255) / 256;
    const int gEw = (E + 7) / 8;                     // wave-per-edge scatter
    const int gM  = (N + 127) / 128;                 // GEMM row blocks
    const int gF  = (N + 1) / 2;                     // finalize: 2 nodes/block

    // --- symmetric normalization ---
    deg_init  <<<gN, 256, 0, stream>>>(deg, N);
    deg_accum <<<gE, 256, 0, stream>>>(ei, E, deg);
    deg_rsqrt <<<gN, 256, 0, stream>>>(deg, N);      // deg -> dinv in place

    // --- layer 1: h1 = relu(GCN(x; c1)) ; z = [h1|x] ---
    gemm_wmma_f32<8,   false><<<gM, 256, 0, stream>>>(x, c1W, nullptr, bufA, N);
    hipMemsetAsync(bufB, 0, (size_t)N * 128 * sizeof(float), stream);
    gcn_scatter<<<gEw, 256, 0, stream>>>(ei, E, deg, bufA, bufB);
    finalize_concat<<<gF, 256, 0, stream>>>(bufB, bufA, deg, c1b, x, bufC, N, 0);

    // --- layer 2: h2 = relu(GCN(z; c2)) ; xr = [x|h2] ---
    gemm_wmma_f32<136, false><<<gM, 256, 0, stream>>>(bufC, c2W, nullptr, bufA, N);
    hipMemsetAsync(bufB, 0, (size_t)N * 128 * sizeof(float), stream);
    gcn_scatter<<<gEw, 256, 0, stream>>>(ei, E, deg, bufA, bufB);
    finalize_concat<<<gF, 256, 0, stream>>>(bufB, bufA, deg, c2b, x, bufC, N, 1);

    // --- head: h4 = relu(xr @ f4W + f4b) ; out = sigmoid([x|h4]@f5W + f5b) ---
    gemm_wmma_f32<136, true><<<gM, 256, 0, stream>>>(bufC, f4W, f4b, bufA, N);
    final_out<<<gN, 256, 0, stream>>>(x, bufA, f5W, f5b, out, N);
}